// HashGrid_36979668418641
// MI455X (gfx1250) — compile-verified
//
#include <hip/hip_runtime.h>
#include <stdint.h>

// ---- Instant-NGP hash-grid constants (match reference) ----
#define HG_T      32768u
#define HG_MASK   (HG_T - 1u)
#define HG_F      2u
#define PRIME_Y   2654435761u
#define PRIME_Z   805459861u

// scale_l = 12*1.5^l - 1 ; res = ceil(scale)+1
#define SCALE0 11.0f
#define SCALE1 17.0f
#define SCALE2 26.0f
#define SCALE3 39.5f
#define RES0   12u
#define RES1   18u
#define RES2   27u
// res^3 <= T for levels 0..2 (dense), level 3 hashed.

// Max dense index actually reachable = res + res*res + res*res*res
// (corner coords go up to res), so touched entries = res*(1+res+res^2)+1.
#define L0_ENTRIES 1885u    // 12*157 + 1
#define L1_ENTRIES 6175u    // 18*343 + 1
// floats staged, rounded up to multiples of 4 for float4/b128 copies
#define L0_FLOATS  3772u
#define L1_FLOATS  12352u
#define L3_FLOATS  (HG_T * 2u)            // 65536
#define L0_OFF     0u
#define L1_OFF     (L0_OFF + L0_FLOATS)   // 3772  (byte off 15088, 8B aligned)
#define L3_OFF     (L1_OFF + L1_FLOATS)   // 16124 (byte off 64496, 8B aligned)
#define LDS_FLOATS (L3_OFF + L3_FLOATS)   // 81660 floats = 326640 B <= 320KB

#define HG_BLOCK 512
#define HG_GRID  1024

typedef int hg_v4i __attribute__((vector_size(16)));
#if defined(__HIP_DEVICE_COMPILE__)
typedef hg_v4i __attribute__((address_space(1))) hg_v4i_global;
typedef hg_v4i __attribute__((address_space(3))) hg_v4i_lds;
#endif

__device__ __forceinline__ float2 hg_lerp2(float2 a, float2 b, float t) {
  float2 r;
  r.x = __builtin_fmaf(t, b.x - a.x, a.x);
  r.y = __builtin_fmaf(t, b.y - a.y, a.y);
  return r;
}

__device__ __forceinline__ float2 hg_trilerp(const float2* __restrict__ t,
    uint32_t i000, uint32_t i100, uint32_t i010, uint32_t i110,
    uint32_t i001, uint32_t i101, uint32_t i011, uint32_t i111,
    float tx, float ty, float tz)
{
  float2 c000 = t[i000], c100 = t[i100], c010 = t[i010], c110 = t[i110];
  float2 c001 = t[i001], c101 = t[i101], c011 = t[i011], c111 = t[i111];
  float2 x00 = hg_lerp2(c000, c100, tx);
  float2 x10 = hg_lerp2(c010, c110, tx);
  float2 x01 = hg_lerp2(c001, c101, tx);
  float2 x11 = hg_lerp2(c011, c111, tx);
  float2 y0  = hg_lerp2(x00, x10, ty);
  float2 y1  = hg_lerp2(x01, x11, ty);
  return hg_lerp2(y0, y1, tz);
}

__device__ __forceinline__ float2 hg_dense(float px, float py, float pz,
    float scale, uint32_t res, const float2* __restrict__ tab)
{
  float fx = __builtin_fmaf(px, scale, 0.5f);
  float fy = __builtin_fmaf(py, scale, 0.5f);
  float fz = __builtin_fmaf(pz, scale, 0.5f);
  float gx = floorf(fx), gy = floorf(fy), gz = floorf(fz);
  float tx = fx - gx, ty = fy - gy, tz = fz - gz;
  uint32_t ix = (uint32_t)gx, iy = (uint32_t)gy, iz = (uint32_t)gz;
  uint32_t r2 = res * res;
  uint32_t b  = ix + iy * res + iz * r2;
  return hg_trilerp(tab,
                    b,            b + 1u,
                    b + res,      b + res + 1u,
                    b + r2,       b + r2 + 1u,
                    b + r2 + res, b + r2 + res + 1u,
                    tx, ty, tz);
}

__device__ __forceinline__ float2 hg_hash(float px, float py, float pz,
    float scale, const float2* __restrict__ tab)
{
  float fx = __builtin_fmaf(px, scale, 0.5f);
  float fy = __builtin_fmaf(py, scale, 0.5f);
  float fz = __builtin_fmaf(pz, scale, 0.5f);
  float gx = floorf(fx), gy = floorf(fy), gz = floorf(fz);
  float tx = fx - gx, ty = fy - gy, tz = fz - gz;
  uint32_t ix = (uint32_t)gx, iy = (uint32_t)gy, iz = (uint32_t)gz;
  uint32_t hx0 = ix,               hx1 = ix + 1u;           // PRIMES[0] == 1
  uint32_t hy0 = iy * PRIME_Y,     hy1 = (iy + 1u) * PRIME_Y;
  uint32_t hz0 = iz * PRIME_Z,     hz1 = (iz + 1u) * PRIME_Z;
  return hg_trilerp(tab,
                    (hx0 ^ hy0 ^ hz0) & HG_MASK, (hx1 ^ hy0 ^ hz0) & HG_MASK,
                    (hx0 ^ hy1 ^ hz0) & HG_MASK, (hx1 ^ hy1 ^ hz0) & HG_MASK,
                    (hx0 ^ hy0 ^ hz1) & HG_MASK, (hx1 ^ hy0 ^ hz1) & HG_MASK,
                    (hx0 ^ hy1 ^ hz1) & HG_MASK, (hx1 ^ hy1 ^ hz1) & HG_MASK,
                    tx, ty, tz);
}

extern "C" __global__ void __launch_bounds__(HG_BLOCK, 1)
hashgrid_enc_kernel(const float* __restrict__ xin,
                    const float* __restrict__ params,
                    float* __restrict__ out,
                    int n)
{
  extern __shared__ float smem[];
  const int tid  = threadIdx.x;
  const int nthr = blockDim.x;

  // ---- Stage L0 / L1 / L3 feature tables into LDS (326,640 B) ----
  {
    const float4* s0 = (const float4*)(params);                        // level 0
    const float4* s1 = (const float4*)(params + HG_T * HG_F);          // level 1
    const float4* s3 = (const float4*)(params + 3u * HG_T * HG_F);     // level 3
    float4* d0 = (float4*)(smem + L0_OFF);
    float4* d1 = (float4*)(smem + L1_OFF);
    float4* d3 = (float4*)(smem + L3_OFF);
    const int n0 = (int)(L0_FLOATS / 4u);
    const int n1 = (int)(L1_FLOATS / 4u);
    const int n3 = (int)(L3_FLOATS / 4u);
#if defined(__gfx1250__) && \
    __has_builtin(__builtin_amdgcn_global_load_async_to_lds_b128) && \
    __has_builtin(__builtin_amdgcn_s_wait_asynccnt)
    // CDNA5 async global->LDS engine (ASYNCcnt), no VGPR round-trip.
    for (int i = tid; i < n0; i += nthr)
      __builtin_amdgcn_global_load_async_to_lds_b128(
          (hg_v4i_global*)(s0 + i), (hg_v4i_lds*)(d0 + i), 0, 0);
    for (int i = tid; i < n1; i += nthr)
      __builtin_amdgcn_global_load_async_to_lds_b128(
          (hg_v4i_global*)(s1 + i), (hg_v4i_lds*)(d1 + i), 0, 0);
    for (int i = tid; i < n3; i += nthr)
      __builtin_amdgcn_global_load_async_to_lds_b128(
          (hg_v4i_global*)(s3 + i), (hg_v4i_lds*)(d3 + i), 0, 0);
    __builtin_amdgcn_s_wait_asynccnt(0);
#else
    for (int i = tid; i < n0; i += nthr) d0[i] = s0[i];
    for (int i = tid; i < n1; i += nthr) d1[i] = s1[i];
    for (int i = tid; i < n3; i += nthr) d3[i] = s3[i];
#endif
    __syncthreads();
  }

  const float2* t0 = (const float2*)(smem + L0_OFF);                  // LDS
  const float2* t1 = (const float2*)(smem + L1_OFF);                  // LDS
  const float2* t3 = (const float2*)(smem + L3_OFF);                  // LDS
  const float2* t2 = (const float2*)(params + 2u * HG_T * HG_F);      // L2-resident

  const int stride = (int)(gridDim.x * blockDim.x);
  for (int i = (int)(blockIdx.x * blockDim.x) + tid; i < n; i += stride) {
    const float* xp = xin + 3u * (uint32_t)i;
    float px = xp[0], py = xp[1], pz = xp[2];

    float2 f0 = hg_dense(px, py, pz, SCALE0, RES0, t0);
    float2 f1 = hg_dense(px, py, pz, SCALE1, RES1, t1);
    float2 f2 = hg_dense(px, py, pz, SCALE2, RES2, t2);
    float2 f3 = hg_hash (px, py, pz, SCALE3, t3);

    float4* o = (float4*)(out + 8u * (uint32_t)i);
    o[0] = make_float4(f0.x, f0.y, f1.x, f1.y);
    o[1] = make_float4(f2.x, f2.y, f3.x, f3.y);
  }
}

extern "C" void kernel_launch(void* const* d_in, const int* in_sizes, int n_in,
                              void* d_out, int out_size, void* d_ws, size_t ws_size,
                              hipStream_t stream) {
  const float* x      = (const float*)d_in[0];
  const float* params = (const float*)d_in[1];
  float* out          = (float*)d_out;
  const int n = in_sizes[0] / 3;

  const size_t shmem = (size_t)LDS_FLOATS * sizeof(float);  // 326,640 B
  // Opt in to >default dynamic LDS (320 KB/WGP on CDNA5). Deterministic,
  // not stream-ordered; safe under graph capture.
  (void)hipFuncSetAttribute((const void*)hashgrid_enc_kernel,
                            hipFuncAttributeMaxDynamicSharedMemorySize,
                            (int)shmem);

  int grid = (n + HG_BLOCK - 1) / HG_BLOCK;
  if (grid > HG_GRID) grid = HG_GRID;
  if (grid < 1) grid = 1;

  hashgrid_enc_kernel<<<grid, HG_BLOCK, shmem, stream>>>(x, params, out, n);
}